// VoxelSampler_62448824484383
// MI455X (gfx1250) — compile-verified
//
#include <hip/hip_runtime.h>
#include <stdint.h>

// VoxelSampler for MI455X (gfx1250, wave32):
//  - dis^2 via V_WMMA_F32_16X16X4_F32 (rank-4 expansion of squared distance)
//  - ballot -> per-ROI bitmask words in LDS
//  - wave-level ordered compaction (popc + shfl prefix scan) = top_k of 0/1 mask

typedef __attribute__((ext_vector_type(2))) float v2f;
typedef __attribute__((ext_vector_type(8))) float v8f;

#define B_      2
#define T_      8
#define R_      128
#define NP_     32768
#define F_      5
#define S_      32
#define RTILE   16
#define NTHREADS 256
#define WAVES   8
#define PASS_PTS 8192
#define NPASS   (NP_ / PASS_PTS)          // 4
#define WORDS_PER_PASS (PASS_PTS / 32)    // 256
#define PTS_PER_WAVE (PASS_PTS / WAVES)   // 1024
#define DTILES  (PTS_PER_WAVE / 32)       // 32 double-tiles of 32 points

#if __has_builtin(__builtin_amdgcn_ballot_w32)
#define BALLOT32(p) __builtin_amdgcn_ballot_w32(p)
#else
#define BALLOT32(p) ((uint32_t)__ballot(p))
#endif

#if __has_builtin(__builtin_amdgcn_wmma_f32_16x16x4_f32)
#define HAVE_WMMA_16X16X4_F32 1
#endif

__global__ __launch_bounds__(NTHREADS) void voxel_sampler_kernel(
    const float* __restrict__ points,   // [B,T,Np,5]
    const float* __restrict__ rois,     // [B,T,R,9]
    float* __restrict__ out)            // [B,R,T*S,5]
{
  __shared__ float4   sA[RTILE];                  // {-2cx, -2cy, cx^2+cy^2, rad^2}
  __shared__ uint32_t sMask[RTILE][WORDS_PER_PASS];
  __shared__ int      sCount[RTILE];
  __shared__ int      sDone;

  const int tid  = threadIdx.x;
  const int lane = tid & 31;
  const int wave = tid >> 5;

  const int wg = blockIdx.x;            // B*T*(R/16) = 128 workgroups
  const int rt = wg & 7;                // roi tile index (R/RTILE = 8)
  const int t  = (wg >> 3) & 7;
  const int b  = wg >> 6;

  const float* __restrict__ pbase = points + (size_t)(b * T_ + t) * NP_ * F_;

  // ---- per-ROI derived params ----
  if (tid < RTILE) {
    const int r = rt * RTILE + tid;
    const float* rp = rois + (size_t)((b * T_ + t) * R_ + r) * 9;
    const float cx = rp[0], cy = rp[1];
    const float dx = rp[3], dy = rp[4];
    const float vx = rp[7], vy = rp[8];
    const float speed = sqrtf(vx * vx + vy * vy);
    float g = powf(1.05f * (1.0f + speed), (float)t * 0.2f);
    g = fminf(g, 2.5f);
    const float rad = sqrtf(dx * dx * 0.25f + dy * dy * 0.25f) * g;
    sA[tid] = make_float4(-2.0f * cx, -2.0f * cy, cx * cx + cy * cy, rad * rad);
    sCount[tid] = 0;
  }
  if (tid == 0) sDone = 0;
  __syncthreads();

  // ---- A-matrix registers (16x4 f32 layout: lanes<16 hold K=0,1; lanes>=16 hold K=2,3)
  const int m = lane & 15;
  v2f Areg;
  if (lane < 16) { Areg.x = sA[m].x; Areg.y = sA[m].y; }
  else           { Areg.x = sA[m].z; Areg.y = 1.0f;    }

  float r2sel[8];
  #pragma unroll
  for (int v = 0; v < 8; ++v)
    r2sel[v] = (lane < 16) ? sA[v].w : sA[v + 8].w;

  for (int pass = 0; pass < NPASS; ++pass) {
    if (sDone) break;                   // uniform across workgroup
    const int passBase = pass * PASS_PTS;

    // ======== Phase 1: WMMA distance tiles -> bitmask words ========
    for (int dt2 = 0; dt2 < DTILES; ++dt2) {
      const int p0 = passBase + wave * PTS_PER_WAVE + dt2 * 32;
      const int w  = wave * DTILES + dt2;     // word index within pass

      v8f d[2];
      #pragma unroll
      for (int h = 0; h < 2; ++h) {
        const int p = p0 + h * 16 + (lane & 15);   // lanes>=16 re-load same 16 pts
        const float px = pbase[(size_t)p * F_ + 0];
        const float py = pbase[(size_t)p * F_ + 1];
        const float s2 = px * px + py * py;
#ifdef HAVE_WMMA_16X16X4_F32
        v2f Breg;                // B 4x16 f32: lanes<16 K=0,1; lanes>=16 K=2,3
        if (lane < 16) { Breg.x = px;   Breg.y = py; }
        else           { Breg.x = 1.0f; Breg.y = s2; }
        v8f c = {};
        d[h] = __builtin_amdgcn_wmma_f32_16x16x4_f32(
            /*neg_a=*/false, Areg, /*neg_b=*/false, Breg,
            /*c_mod=*/(short)0, c, /*reuse_a=*/false, /*reuse_b=*/false);
#else
        #pragma unroll
        for (int v = 0; v < 8; ++v) {
          const int mr = v + ((lane >= 16) ? 8 : 0);
          const float4 A4 = sA[mr];
          d[h][v] = fmaf(A4.x, px, fmaf(A4.y, py, s2 + A4.z));
        }
#endif
      }

      #pragma unroll
      for (int v = 0; v < 8; ++v) {
        const uint32_t bal0 = BALLOT32(d[0][v] <= r2sel[v]);
        const uint32_t bal1 = BALLOT32(d[1][v] <= r2sel[v]);
        if (lane == 0) {
          sMask[v][w]     = (bal0 & 0xFFFFu) | (bal1 << 16);          // roi v
          sMask[v + 8][w] = (bal0 >> 16)     | (bal1 & 0xFFFF0000u);  // roi v+8
        }
      }
    }
    __syncthreads();

    // ======== Phase 2: ordered compaction (2 ROIs per wave) ========
    for (int k = 0; k < 2; ++k) {
      const int rr = wave + 8 * k;
      int cnt = sCount[rr];
      if (cnt < S_) {
        const int rg = rt * RTILE + rr;
        float* __restrict__ obase =
            out + (size_t)((b * R_ + rg) * (T_ * S_) + t * S_) * F_;
        for (int cw = 0; cw < WORDS_PER_PASS; cw += 32) {
          const uint32_t mword = sMask[rr][cw + lane];
          const int c = __popc(mword);
          int inc = c;
          #pragma unroll
          for (int off = 1; off < 32; off <<= 1) {
            const int nv = __shfl_up(inc, off, 32);
            if (lane >= off) inc += nv;
          }
          const int excl  = inc - c;
          const int total = __shfl(inc, 31, 32);
          int slot = cnt + excl;
          uint32_t mm = mword;
          while (mm && slot < S_) {
            const int bit = __ffs((int)mm) - 1;
            mm &= mm - 1;
            const int pidx = passBase + (cw + lane) * 32 + bit;
            const float* sp = pbase + (size_t)pidx * F_;
            float* op = obase + (size_t)slot * F_;
            op[0] = sp[0]; op[1] = sp[1]; op[2] = sp[2];
            op[3] = sp[3]; op[4] = sp[4];
            ++slot;
          }
          cnt += total;
          if (cnt >= S_) break;         // wave-uniform
        }
        if (lane == 0) sCount[rr] = cnt < S_ ? cnt : S_;
      }
    }
    __syncthreads();
    if (tid == 0) {
      int done = 1;
      for (int i = 0; i < RTILE; ++i) if (sCount[i] < S_) done = 0;
      sDone = done;
    }
    __syncthreads();
  }

  // ======== Phase 3: zero-fill unused slots ========
  for (int task = tid; task < RTILE * S_; task += NTHREADS) {
    const int rr = task >> 5;
    const int s  = task & 31;
    if (s >= sCount[rr]) {
      const int rg = rt * RTILE + rr;
      float* op = out + (size_t)((b * R_ + rg) * (T_ * S_) + t * S_ + s) * F_;
      op[0] = 0.0f; op[1] = 0.0f; op[2] = 0.0f; op[3] = 0.0f; op[4] = 0.0f;
    }
  }
}

extern "C" void kernel_launch(void* const* d_in, const int* in_sizes, int n_in,
                              void* d_out, int out_size, void* d_ws, size_t ws_size,
                              hipStream_t stream) {
  (void)in_sizes; (void)n_in; (void)d_ws; (void)ws_size; (void)out_size;
  const float* points = (const float*)d_in[0];        // [2,8,32768,5] f32
  const float* rois   = (const float*)d_in[1];        // [2,8,128,9]   f32
  // d_in[2] = num_sample (device scalar) -- fixed at 32 per reference setup.
  float* out = (float*)d_out;                          // [2,128,256,5] f32
  dim3 grid(B_ * T_ * (R_ / RTILE));                   // 128 workgroups
  dim3 block(NTHREADS);                                // 8 wave32s
  voxel_sampler_kernel<<<grid, block, 0, stream>>>(points, rois, out);
}